// TransferNBFNet_90692529422648
// MI455X (gfx1250) — compile-verified
//
#include <hip/hip_runtime.h>
#include <hip/hip_bf16.h>

typedef __attribute__((ext_vector_type(2))) float v2f;
typedef __attribute__((ext_vector_type(8))) float v8f;

#define N_NODE 20000
#define E_EDGE 80000
#define BS 8
#define DD 32
#define LAY 6
#define KTOT 416               // 13*d
#define ELEM (N_NODE*BS*DD)    // 5,120,000 floats
#define ROWS (N_NODE*BS)       // 160,000

#if __has_builtin(__builtin_amdgcn_global_load_async_to_lds_b32)
#define HAVE_ASYNC_LDS 1
typedef __attribute__((address_space(1))) int gas_int;  // global addrspace
typedef __attribute__((address_space(3))) int las_int;  // LDS addrspace
#else
#define HAVE_ASYNC_LDS 0
#endif

// ---- monotonic float <-> uint encoding for atomic max/min on floats ----
__device__ __forceinline__ unsigned encf(float f){
  unsigned u = __float_as_uint(f);
  return (u & 0x80000000u) ? ~u : (u | 0x80000000u);
}
__device__ __forceinline__ float decf(unsigned k){
  unsigned u = (k & 0x80000000u) ? (k & 0x7fffffffu) : ~k;
  return __uint_as_float(u);
}

// ---- prep: is_t_neg / h0 / r0 / t_new / query ----
__global__ void k_prep(const int* __restrict__ h_index, const int* __restrict__ t_index,
                       const int* __restrict__ r_index, const float* __restrict__ rel_query,
                       int* __restrict__ tnew, int* __restrict__ hr,
                       float* __restrict__ query, float* __restrict__ redbuf){
  int t = threadIdx.x;
  if (t < BS){
    int b = t;
    int h0v = h_index[b*33];
    bool isneg = true;
    for (int j=0;j<33;j++) isneg = isneg && (h_index[b*33+j]==h0v);
    int h0 = isneg ? h_index[b*33] : t_index[b*33];
    int r0 = isneg ? r_index[b*33] : r_index[b*33] + 64;
    hr[b] = h0; hr[BS+b] = r0;
    for (int j=0;j<33;j++) tnew[b*33+j] = isneg ? t_index[b*33+j] : h_index[b*33+j];
  }
  if (t==0) redbuf[0] = 0.f;
  __syncthreads();
  int b = t>>5, c = t&31;                 // 8*32 = 256 threads exactly
  query[b*DD+c] = rel_query[hr[BS+b]*DD + c];
}

// ---- zero boundary/x, deg=1 ----
__global__ void k_fill(float* __restrict__ boundary, float* __restrict__ xg, float* __restrict__ deg){
  long long i = (long long)blockIdx.x*256 + threadIdx.x;
  if (i < ELEM){ boundary[i]=0.f; xg[i]=0.f; }
  if (i < N_NODE) deg[i]=1.f;
}

// ---- boundary[h0[b], b, :] = query[b,:]  (and x starts as boundary) ----
__global__ void k_scatter(const int* __restrict__ hr, const float* __restrict__ query,
                          float* __restrict__ boundary, float* __restrict__ xg){
  int t = threadIdx.x; int b = t>>5, c = t&31;
  float v = query[b*DD+c];
  long long idx = ((long long)hr[b]*BS + b)*DD + c;
  boundary[idx] = v; xg[idx] = v;
}

// ---- degree accumulation over concatenated dst = [edge_dst, edge_src] ----
__global__ void k_deg(const int* __restrict__ es, const int* __restrict__ ed, float* __restrict__ deg){
  int e = blockIdx.x*256 + threadIdx.x;
  if (e < 2*E_EDGE){
    int dt = (e < E_EDGE) ? ed[e] : es[e-E_EDGE];
    atomicAdd(&deg[dt], 1.f);
  }
}

// ---- sum of log(deg) ----
__global__ void k_logred(const float* __restrict__ deg, float* __restrict__ redbuf){
  __shared__ float sm[256];
  int i = blockIdx.x*256 + threadIdx.x;
  sm[threadIdx.x] = (i < N_NODE) ? logf(deg[i]) : 0.f;
  __syncthreads();
  for (int s=128;s>0;s>>=1){ if (threadIdx.x<s) sm[threadIdx.x]+=sm[threadIdx.x+s]; __syncthreads(); }
  if (threadIdx.x==0) atomicAdd(redbuf, sm[0]);
}

// ---- per-node scale triple + 1/deg ----
__global__ void k_scales(const float* __restrict__ deg, const float* __restrict__ redbuf,
                         float* __restrict__ scl){
  int n = blockIdx.x*256 + threadIdx.x;
  if (n < N_NODE){
    float mean = redbuf[0] * (1.f/(float)N_NODE);
    float s = logf(deg[n]) / mean;
    scl[n*4+0] = 1.f;
    scl[n*4+1] = s;
    scl[n*4+2] = 1.f / fmaxf(s, 0.01f);
    scl[n*4+3] = 1.f / deg[n];
  }
}

// ---- rel = rel_query(128x32) @ Wrel[i](32x32) + brel[i] (tiny) ----
__global__ void k_rel(const float* __restrict__ rel_query, const float* __restrict__ Wrel,
                      const float* __restrict__ brel, float* __restrict__ rel, int layer){
  int t = threadIdx.x;
  for (int o=t;o<128*DD;o+=256){
    int r=o>>5, c=o&31;
    float acc = brel[layer*DD+c];
    for (int k=0;k<DD;k++) acc += rel_query[r*DD+k]*Wrel[(layer*DD+k)*DD + c];
    rel[o]=acc;
  }
}

// ---- init segment accumulators ----
__global__ void k_accinit(float* __restrict__ asum, float* __restrict__ asq,
                          unsigned* __restrict__ amax, unsigned* __restrict__ amin){
  long long i = (long long)blockIdx.x*256 + threadIdx.x;
  if (i < ELEM){
    asum[i]=0.f; asq[i]=0.f;
    amax[i]=encf(-3.0e38f); amin[i]=encf(3.0e38f);
  }
}

// ---- edge messages + segment sum/sumsq/max/min (one edge per block, 256 = bs*d lanes) ----
__global__ __launch_bounds__(256) void k_msg(
    const int* __restrict__ es, const int* __restrict__ ed, const int* __restrict__ et,
    const float* __restrict__ xg, const float* __restrict__ rel,
    float* __restrict__ asum, float* __restrict__ asq,
    unsigned* __restrict__ amax, unsigned* __restrict__ amin){
  int e = blockIdx.x;
  int t = threadIdx.x;
  int s, dt, ty;
  if (e < E_EDGE){ s = es[e]; dt = ed[e]; ty = et[e]; }
  else           { s = ed[e-E_EDGE]; dt = es[e-E_EDGE]; ty = et[e-E_EDGE] + 64; }
  float m = xg[(long long)s*256 + t] * rel[ty*DD + (t&31)];
  long long idx = (long long)dt*256 + t;
  atomicAdd(&asum[idx], m);
  atomicAdd(&asq[idx], m*m);
  atomicMax(&amax[idx], encf(m));
  atomicMin(&amin[idx], encf(m));
}

// ---- fused: feats -> (x||update) @ Wlin[i] via V_WMMA_F32_16X16X4_F32 -> LN -> ReLU -> residual ----
__global__ __launch_bounds__(256) void k_layer(
    float* __restrict__ xg,
    const float* __restrict__ asum, const float* __restrict__ asq,
    const unsigned* __restrict__ amax, const unsigned* __restrict__ amin,
    const float* __restrict__ boundary, const float* __restrict__ scl,
    const float* __restrict__ Wlin, const float* __restrict__ blin,
    const float* __restrict__ lng, const float* __restrict__ lnb, int layer){
  __shared__ float xs[64][DD];    // 8 KB   : x rows (A k<32, residual)
  __shared__ float fs[64][128];   // 32 KB  : feats (mean,max,min,std interleaved)
  __shared__ float wc[104][DD];   // 13 KB  : Wlin K-chunk
  __shared__ float os[64][DD];    // 8 KB   : GEMM output tile
  __shared__ float sc[8][4];      // node scales {1, s, 1/clip(s), 1/deg}

  int t = threadIdx.x;
  int row0 = blockIdx.x * 64;     // rows are (node*BS + b)

  if (t < 32){ int nn=t>>2, c=t&3; sc[nn][c] = scl[(row0/8 + nn)*4 + c]; }
  for (int o=t;o<64*DD;o+=256){ int r=o>>5,c=o&31; xs[r][c] = xg[(long long)(row0+r)*DD + c]; }
  for (int o=t;o<64*DD;o+=256){
    int r=o>>5, kd=o&31;
    long long gi = (long long)(row0+r)*DD + kd;
    float bv   = boundary[gi];
    float dinv = scl[((row0+r)>>3)*4 + 3];
    float mean = (asum[gi] + bv)      * dinv;
    float sqm  = (asq[gi]  + bv*bv)   * dinv;
    float mx   = fmaxf(decf(amax[gi]), bv);
    float mn   = fminf(decf(amin[gi]), bv);
    float sd   = sqrtf(fmaxf(sqm - mean*mean, 1e-6f));
    fs[r][kd*4+0]=mean; fs[r][kd*4+1]=mx; fs[r][kd*4+2]=mn; fs[r][kd*4+3]=sd;
  }
  __syncthreads();

  int wave = t>>5, lane = t&31;
  int m0 = (wave>>1)*16, n0 = (wave&1)*16;     // 4x2 wave tiles = 64x32 block tile
  int half = lane>>4, l16 = lane&15;

  v8f acc; for (int v=0;v<8;v++) acc[v]=0.f;

  // A(r,k): k<32 -> x ; else update[u]=feats[u/3]*scale[u%3]
  auto aval = [&](int r, int k)->float {
    if (k < DD) return xs[r][k];
    int u = k - DD; int f = u/3; int s = u - 3*f;
    return fs[r][f] * sc[r>>3][s];
  };

  for (int kc=0;kc<4;kc++){
    int kb = kc*104;
#if HAVE_ASYNC_LDS
    // gfx1250 async cache->LDS transfer, tracked by ASYNCcnt
    for (int o=t;o<104*DD;o+=256){
      int r=o>>5, c=o&31;
      __builtin_amdgcn_global_load_async_to_lds_b32(
          (gas_int*)&Wlin[((long long)layer*KTOT + kb + r)*DD + c],
          (las_int*)&wc[r][c],
          0, 0);
    }
  #if __has_builtin(__builtin_amdgcn_s_wait_asynccnt)
    __builtin_amdgcn_s_wait_asynccnt(0);
  #else
    asm volatile("s_wait_asynccnt 0x0" ::: "memory");
  #endif
#else
    for (int o=t;o<104*DD;o+=256){
      int r=o>>5, c=o&31;
      wc[r][c] = Wlin[((long long)layer*KTOT + kb + r)*DD + c];
    }
#endif
    __syncthreads();
    for (int kk=0;kk<104;kk+=4){
      int klane = kk + 2*half;           // lanes 0-15: K0/K1, lanes 16-31: K2/K3
      int ar = m0 + l16;
      v2f a, bf;
      a[0]  = aval(ar, kb + klane);
      a[1]  = aval(ar, kb + klane + 1);
      bf[0] = wc[klane    ][n0 + l16];
      bf[1] = wc[klane + 1][n0 + l16];
      acc = __builtin_amdgcn_wmma_f32_16x16x4_f32(
              /*neg_a=*/false, a, /*neg_b=*/false, bf,
              /*c_mod=*/(short)0, acc, /*reuse_a=*/false, /*reuse_b=*/false);
    }
    __syncthreads();
  }

  // C layout: acc[v] = D[m0 + v + 8*half][n0 + l16]
  for (int v=0;v<8;v++) os[m0 + v + 8*half][n0 + l16] = acc[v] + blin[layer*DD + n0 + l16];
  __syncthreads();

  // LayerNorm(d=32) + ReLU + residual, one row per thread
  if (t < 64){
    float mu = 0.f;
    for (int c=0;c<DD;c++) mu += os[t][c];
    mu *= (1.f/DD);
    float var = 0.f;
    for (int c=0;c<DD;c++){ float d_ = os[t][c]-mu; var += d_*d_; }
    var *= (1.f/DD);
    float inv = rsqrtf(var + 1e-5f);
    for (int c=0;c<DD;c++){
      float y = (os[t][c]-mu)*inv*lng[layer*DD+c] + lnb[layer*DD+c];
      y = fmaxf(y, 0.f) + xs[t][c];
      xg[(long long)(row0+t)*DD + c] = y;
    }
  }
}

// ---- final MLP scoring over (bs, 33) gathered nodes ----
__global__ void k_final(const float* __restrict__ xg, const float* __restrict__ query,
                        const int* __restrict__ tnew,
                        const float* __restrict__ W1, const float* __restrict__ b1,
                        const float* __restrict__ W2, const float* __restrict__ b2,
                        float* __restrict__ out){
  int t = threadIdx.x;
  for (int o=t;o<BS*33;o+=256){
    int b = o/33;
    int node = tnew[o];
    float f[64];
    for (int k=0;k<DD;k++) f[k]    = xg[((long long)node*BS + b)*DD + k];
    for (int k=0;k<DD;k++) f[DD+k] = query[b*DD + k];
    float score = b2[0];
    for (int h=0;h<64;h++){
      float acc = b1[h];
      for (int k=0;k<64;k++) acc += f[k]*W1[k*64+h];
      score += fmaxf(acc, 0.f) * W2[h];
    }
    out[o] = score;
  }
}

extern "C" void kernel_launch(void* const* d_in, const int* in_sizes, int n_in,
                              void* d_out, int out_size, void* d_ws, size_t ws_size,
                              hipStream_t stream){
  const int*   edge_src  = (const int*)  d_in[0];
  const int*   edge_dst  = (const int*)  d_in[1];
  const int*   edge_type = (const int*)  d_in[2];
  const int*   h_index   = (const int*)  d_in[3];
  const int*   t_index   = (const int*)  d_in[4];
  const int*   r_index   = (const int*)  d_in[5];
  // d_in[6] = num_node (device scalar; topology constants compiled in)
  const float* rel_query = (const float*)d_in[7];
  const float* Wrel      = (const float*)d_in[8];
  const float* brel      = (const float*)d_in[9];
  const float* Wlin      = (const float*)d_in[10];
  const float* blin      = (const float*)d_in[11];
  const float* ln_g      = (const float*)d_in[12];
  const float* ln_b      = (const float*)d_in[13];
  const float* W1        = (const float*)d_in[14];
  const float* b1        = (const float*)d_in[15];
  const float* W2        = (const float*)d_in[16];
  const float* b2        = (const float*)d_in[17];
  float* out = (float*)d_out;

  float* p = (float*)d_ws;
  float*    boundary = p; p += ELEM;
  float*    xg       = p; p += ELEM;
  float*    asum     = p; p += ELEM;
  float*    asq      = p; p += ELEM;
  unsigned* amax     = (unsigned*)p; p += ELEM;
  unsigned* amin     = (unsigned*)p; p += ELEM;
  float*    rel      = p; p += 128*DD;
  float*    query    = p; p += BS*DD;
  float*    deg      = p; p += N_NODE;
  float*    scl      = p; p += N_NODE*4;
  float*    redbuf   = p; p += 4;
  int*      tnew     = (int*)p; p += BS*33;
  int*      hr       = (int*)p; p += 16;

  int gElem = (ELEM + 255)/256;
  k_prep   <<<1, 256, 0, stream>>>(h_index, t_index, r_index, rel_query, tnew, hr, query, redbuf);
  k_fill   <<<gElem, 256, 0, stream>>>(boundary, xg, deg);
  k_scatter<<<1, 256, 0, stream>>>(hr, query, boundary, xg);
  k_deg    <<<(2*E_EDGE+255)/256, 256, 0, stream>>>(edge_src, edge_dst, deg);
  k_logred <<<(N_NODE+255)/256, 256, 0, stream>>>(deg, redbuf);
  k_scales <<<(N_NODE+255)/256, 256, 0, stream>>>(deg, redbuf, scl);

  for (int i=0;i<LAY;i++){
    k_rel    <<<1, 256, 0, stream>>>(rel_query, Wrel, brel, rel, i);
    k_accinit<<<gElem, 256, 0, stream>>>(asum, asq, amax, amin);
    k_msg    <<<2*E_EDGE, 256, 0, stream>>>(edge_src, edge_dst, edge_type, xg, rel,
                                            asum, asq, amax, amin);
    k_layer  <<<ROWS/64, 256, 0, stream>>>(xg, asum, asq, amax, amin, boundary, scl,
                                           Wlin, blin, ln_g, ln_b, i);
  }
  k_final<<<1, 256, 0, stream>>>(xg, query, tnew, W1, b1, W2, b2, out);
}